// FeatureAttentionModule_3985729650937
// MI455X (gfx1250) — compile-verified
//
#include <hip/hip_runtime.h>

// FeatureAttentionModule forward for MI455X (gfx1250), wave32 + f32 WMMA.
// One workgroup per batch element; activations LDS-resident across all 6 layers.
// Each wave computes 2-3 adjacent 16x16 N-tiles per task: A fragments are
// shared and the independent accumulator chains hide WMMA RAW latency
// (1 block per WGP due to ~169KB LDS, so per-wave ILP is the lever).

#define N_EMB   1000000
#define B_SZ    1024
#define L_SEQ   50
#define S_TOK   81       // real tokens
#define S_PAD   96       // padded to 6 x 16
#define DIM     64
#define SX      68       // LDS row stride for [S_PAD x 64] buffers (bank-skewed)
#define SS      100      // LDS row stride for the [S_PAD x S_PAD] score buffer
#define NLAYER  6
#define NEXP    5
#define NTHREADS 256
#define NW      8        // wave32 waves per block

typedef __attribute__((ext_vector_type(2))) float v2f;
typedef __attribute__((ext_vector_type(8))) float v8f;

// ---------------- wave32 reductions ----------------
__device__ __forceinline__ float wred_sum(float v) {
#pragma unroll
  for (int m = 16; m >= 1; m >>= 1) v += __shfl_xor(v, m, 32);
  return v;
}
__device__ __forceinline__ float wred_max(float v) {
#pragma unroll
  for (int m = 16; m >= 1; m >>= 1) v = fmaxf(v, __shfl_xor(v, m, 32));
  return v;
}

// ---------------- f32 WMMA 16x16x4 ----------------
__device__ __forceinline__ v8f wmma4(v2f a, v2f b, v8f c) {
  // D = A(16x4) * B(4x16) + C(16x16), all f32.
  return __builtin_amdgcn_wmma_f32_16x16x4_f32(
      /*neg_a=*/false, a, /*neg_b=*/false, b,
      /*c_mod=*/(short)0, c, /*reuse_a=*/false, /*reuse_b=*/false);
}

// A fragment (ISA 7.12.2, 32-bit A 16x4): lane l -> row mBase + l%16,
// VGPR j -> K = k0 + (l/16)*2 + j.  src is row-major with given stride.
__device__ __forceinline__ v2f frag_rowpair(const float* src, int stride,
                                            int mBase, int k0, int lane) {
  const float* p = src + (mBase + (lane & 15)) * stride + k0 + ((lane >> 4) << 1);
  v2f r; r.x = p[0]; r.y = p[1];
  return r;
}
// B fragment when B is stored directly as K x N row-major:
// lane l -> col nBase + l%16, VGPR j -> K = k0 + (l/16)*2 + j.
__device__ __forceinline__ v2f frag_colpair(const float* src, int stride,
                                            int nBase, int k0, int lane) {
  const float* p = src + (k0 + ((lane >> 4) << 1)) * stride + nBase + (lane & 15);
  v2f r; r.x = p[0]; r.y = p[stride];
  return r;
}

// acc[j] += A[M x K] * W^T   (W stored [N x K] row-major; NT adjacent 16-col tiles)
template <int KTOT, int NT>
__device__ __forceinline__ void gemm_awt_multi(const float* A, int sa, int mB,
                                               const float* W, int sw, int nB,
                                               int lane, v8f* acc) {
#pragma unroll
  for (int k0 = 0; k0 < KTOT; k0 += 4) {
    v2f a = frag_rowpair(A, sa, mB, k0, lane);
#pragma unroll
    for (int j = 0; j < NT; ++j)
      acc[j] = wmma4(a, frag_rowpair(W, sw, nB + j * 16, k0, lane), acc[j]);
  }
}
// acc[j] += A[M x K] * B[K x N]  (B row-major; NT adjacent 16-col tiles)
template <int KTOT, int NT>
__device__ __forceinline__ void gemm_akn_multi(const float* A, int sa, int mB,
                                               const float* Bm, int sb, int nB,
                                               int lane, v8f* acc) {
#pragma unroll
  for (int k0 = 0; k0 < KTOT; k0 += 4) {
    v2f a = frag_rowpair(A, sa, mB, k0, lane);
#pragma unroll
    for (int j = 0; j < NT; ++j)
      acc[j] = wmma4(a, frag_colpair(Bm, sb, nB + j * 16, k0, lane), acc[j]);
  }
}

// Wave-per-row LayerNorm over 64 features (lane holds cols lane, lane+32).
__device__ __forceinline__ void layernorm_rows(const float* src, float* dst,
                                               const float* g, const float* b,
                                               int lane, int wave) {
  for (int s = wave; s < S_PAD; s += NW) {
    float a0 = src[s * SX + lane];
    float a1 = src[s * SX + lane + 32];
    float mu = wred_sum(a0 + a1) * (1.0f / 64.0f);
    float d0 = a0 - mu, d1 = a1 - mu;
    float var = wred_sum(d0 * d0 + d1 * d1) * (1.0f / 64.0f);
    float rs = rsqrtf(var + 1e-5f);
    dst[s * SX + lane]      = d0 * rs * g[lane]      + b[lane];
    dst[s * SX + lane + 32] = d1 * rs * g[lane + 32] + b[lane + 32];
  }
}

__global__ __launch_bounds__(NTHREADS)
void fam_forward(const int* __restrict__ x_cat, const float* __restrict__ x_num,
                 const float* __restrict__ x_eng,
                 const float* __restrict__ emb_table,
                 const float* __restrict__ emb_eng_table,
                 const float* __restrict__ emb_bias,
                 const float* __restrict__ eng_bias,
                 const float* __restrict__ ln_eng_g, const float* __restrict__ ln_eng_b,
                 const float* __restrict__ ln1_g, const float* __restrict__ ln1_b,
                 const float* __restrict__ ln2_g, const float* __restrict__ ln2_b,
                 const float* __restrict__ attn_in_w, const float* __restrict__ attn_in_b,
                 const float* __restrict__ attn_out_w, const float* __restrict__ attn_out_b,
                 const float* __restrict__ gate_w, const float* __restrict__ gate_b,
                 const float* __restrict__ moe_w1, const float* __restrict__ moe_b1,
                 const float* __restrict__ moe_w2, const float* __restrict__ moe_b2,
                 float* __restrict__ out) {
  extern __shared__ float smem[];
  float* xs = smem;                    // residual stream [S_PAD][SX]
  float* xn = smem + 1 * S_PAD * SX;   // normalized x
  float* qs = smem + 2 * S_PAD * SX;   // q  (later: attn output o, MoE h)
  float* ks = smem + 3 * S_PAD * SX;   // k
  float* vs = smem + 4 * S_PAD * SX;   // v
  float* sc = smem + 5 * S_PAD * SX;   // scores [S_PAD][SS] (later: gates)
  float* hs = qs;                      // MoE hidden aliases q
  float* gt = sc;                      // gates alias score buffer

  const int b    = blockIdx.x;
  const int tid  = threadIdx.x;
  const int lane = tid & 31;
  const int wave = tid >> 5;

  // ---------------- embedding prologue ----------------
  for (int idx = tid; idx < S_PAD * DIM; idx += NTHREADS) {
    int row = idx >> 6, d = idx & 63;
    float v;
    if (row < 32) {                       // categorical embeddings (last step)
      int ci = x_cat[(b * L_SEQ + (L_SEQ - 1)) * 32 + row];
      v = emb_table[ci * DIM + d] + emb_bias[row * DIM + d];
    } else if (row < 64) {                // numeric-feature embeddings
      int j = row - 32;
      int ni = N_EMB - 32 + 1 + j;
      v = emb_table[ni * DIM + d] * x_num[(b * L_SEQ + (L_SEQ - 1)) * 32 + j]
          + emb_bias[row * DIM + d];
    } else if (row < 80) {                // engineered features
      int j = row - 64;
      v = emb_eng_table[(1 + j) * DIM + d] * x_eng[b * 16 + j] + eng_bias[j * DIM + d];
    } else if (row == 80) {               // dummy token = emb_eng_table[0]
      v = emb_eng_table[d];
    } else {                              // pad rows: keep finite
      v = 0.0f;
    }
    xs[row * SX + d] = v;
  }
  __syncthreads();
  layernorm_rows(xs, xs, ln_eng_g, ln_eng_b, lane, wave);
  __syncthreads();

  // ---------------- transformer layers ----------------
  for (int li = 0; li < NLAYER; ++li) {
    const float* g1   = ln1_g + li * DIM;
    const float* b1l  = ln1_b + li * DIM;
    const float* g2   = ln2_g + li * DIM;
    const float* b2l  = ln2_b + li * DIM;
    const float* Wqkv = attn_in_w  + li * 3 * DIM * DIM;
    const float* Bqkv = attn_in_b  + li * 3 * DIM;
    const float* Wout = attn_out_w + li * DIM * DIM;
    const float* Bout = attn_out_b + li * DIM;
    const float* Gw   = gate_w + li * NEXP * DIM;
    const float* Gb   = gate_b + li * NEXP;

    // LN1 -> xn
    layernorm_rows(xs, xn, g1, b1l, lane, wave);
    __syncthreads();

    // QKV = xn @ Wqkv^T + b : 24 tasks = (mt 0..5) x (ng 0..3, 3 n-tiles each)
    for (int t = wave; t < 24; t += NW) {
      int mt = t >> 2, ng = t & 3;
      v8f acc[3] = {};
      gemm_awt_multi<DIM, 3>(xn, SX, mt * 16, Wqkv, DIM, ng * 48, lane, acc);
#pragma unroll
      for (int j = 0; j < 3; ++j) {
        int nt = ng * 3 + j;
        int   col  = (nt & 3) * 16 + (lane & 15);
        float bias = Bqkv[nt * 16 + (lane & 15)];
        float* dst = (nt < 4) ? qs : (nt < 8) ? ks : vs;
        bool  isV  = (nt >= 8);
#pragma unroll
        for (int r = 0; r < 8; ++r) {
          int row = mt * 16 + r + ((lane >> 4) << 3);
          float v = acc[j][r] + bias;
          if (isV && row >= S_TOK) v = 0.0f;   // pad tokens contribute 0 to attn@V
          dst[row * SX + col] = v;
        }
      }
    }
    __syncthreads();

    // scores = q @ k^T * 1/sqrt(D) : 12 tasks = (mt 0..5) x (ng 0..1, 3 tiles)
    for (int t = wave; t < 12; t += NW) {
      int mt = t >> 1, ng = t & 1;
      v8f acc[3] = {};
      gemm_awt_multi<DIM, 3>(qs, SX, mt * 16, ks, SX, ng * 48, lane, acc);
#pragma unroll
      for (int j = 0; j < 3; ++j) {
        int col = (ng * 3 + j) * 16 + (lane & 15);
#pragma unroll
        for (int r = 0; r < 8; ++r) {
          int row = mt * 16 + r + ((lane >> 4) << 3);
          sc[row * SS + col] = (col < S_TOK) ? acc[j][r] * 0.125f : -1e30f;
        }
      }
    }
    __syncthreads();

    // row softmax (96 cols = 3 per lane); zero pad rows
    for (int s = wave; s < S_PAD; s += NW) {
      float v0 = sc[s * SS + lane];
      float v1 = sc[s * SS + lane + 32];
      float v2 = sc[s * SS + lane + 64];
      if (s < S_TOK) {
        float m  = wred_max(fmaxf(v0, fmaxf(v1, v2)));
        float e0 = __expf(v0 - m), e1 = __expf(v1 - m), e2 = __expf(v2 - m);
        float inv = 1.0f / wred_sum(e0 + e1 + e2);
        sc[s * SS + lane]      = e0 * inv;
        sc[s * SS + lane + 32] = e1 * inv;
        sc[s * SS + lane + 64] = e2 * inv;
      } else {
        sc[s * SS + lane] = 0.0f;
        sc[s * SS + lane + 32] = 0.0f;
        sc[s * SS + lane + 64] = 0.0f;
      }
    }
    __syncthreads();

    // o = attn @ v -> overwrite qs : 12 tasks = (mt 0..5) x (ng 0..1, 2 tiles)
    for (int t = wave; t < 12; t += NW) {
      int mt = t >> 1, ng = t & 1;
      v8f acc[2] = {};
      gemm_akn_multi<S_PAD, 2>(sc, SS, mt * 16, vs, SX, ng * 32, lane, acc);
#pragma unroll
      for (int j = 0; j < 2; ++j) {
        int col = (ng * 2 + j) * 16 + (lane & 15);
#pragma unroll
        for (int r = 0; r < 8; ++r) {
          int row = mt * 16 + r + ((lane >> 4) << 3);
          qs[row * SX + col] = acc[j][r];
        }
      }
    }
    __syncthreads();

    // x += o @ Wout^T + bout : 12 tasks, 2 tiles each
    for (int t = wave; t < 12; t += NW) {
      int mt = t >> 1, ng = t & 1;
      v8f acc[2] = {};
      gemm_awt_multi<DIM, 2>(qs, SX, mt * 16, Wout, DIM, ng * 32, lane, acc);
#pragma unroll
      for (int j = 0; j < 2; ++j) {
        int   col  = (ng * 2 + j) * 16 + (lane & 15);
        float bias = Bout[(ng * 2 + j) * 16 + (lane & 15)];
#pragma unroll
        for (int r = 0; r < 8; ++r) {
          int row = mt * 16 + r + ((lane >> 4) << 3);
          xs[row * SX + col] += acc[j][r] + bias;
        }
      }
    }
    __syncthreads();

    // LN2 -> xn
    layernorm_rows(xs, xn, g2, b2l, lane, wave);
    __syncthreads();

    // gate softmax over 5 experts (one thread per row)
    if (tid < S_PAD) {
      float lg[NEXP];
      float mx = -3.4e38f;
      for (int e = 0; e < NEXP; ++e) {
        float s = Gb[e];
        for (int d = 0; d < DIM; ++d) s += xn[tid * SX + d] * Gw[e * DIM + d];
        lg[e] = s;
        mx = fmaxf(mx, s);
      }
      float ssum = 0.0f;
      for (int e = 0; e < NEXP; ++e) { lg[e] = __expf(lg[e] - mx); ssum += lg[e]; }
      float inv = 1.0f / ssum;
      for (int e = 0; e < NEXP; ++e) gt[tid * 8 + e] = lg[e] * inv;
    }
    __syncthreads();

    // dense MoE: for each expert, h = relu(xn@W1+b1); x += gate_e * (h@W2+b2)
    for (int e = 0; e < NEXP; ++e) {
      const float* W1 = moe_w1 + (li * NEXP + e) * DIM * DIM;
      const float* B1 = moe_b1 + (li * NEXP + e) * DIM;
      const float* W2 = moe_w2 + (li * NEXP + e) * DIM * DIM;
      const float* B2 = moe_b2 + (li * NEXP + e) * DIM;

      for (int t = wave; t < 12; t += NW) {
        int mt = t >> 1, ng = t & 1;
        v8f acc[2] = {};
        gemm_akn_multi<DIM, 2>(xn, SX, mt * 16, W1, DIM, ng * 32, lane, acc);
#pragma unroll
        for (int j = 0; j < 2; ++j) {
          int   col  = (ng * 2 + j) * 16 + (lane & 15);
          float bias = B1[(ng * 2 + j) * 16 + (lane & 15)];
#pragma unroll
          for (int r = 0; r < 8; ++r) {
            int row = mt * 16 + r + ((lane >> 4) << 3);
            hs[row * SX + col] = fmaxf(acc[j][r] + bias, 0.0f);
          }
        }
      }
      __syncthreads();

      for (int t = wave; t < 12; t += NW) {
        int mt = t >> 1, ng = t & 1;
        v8f acc[2] = {};
        gemm_akn_multi<DIM, 2>(hs, SX, mt * 16, W2, DIM, ng * 32, lane, acc);
#pragma unroll
        for (int j = 0; j < 2; ++j) {
          int   col  = (ng * 2 + j) * 16 + (lane & 15);
          float bias = B2[(ng * 2 + j) * 16 + (lane & 15)];
#pragma unroll
          for (int r = 0; r < 8; ++r) {
            int row = mt * 16 + r + ((lane >> 4) << 3);
            xs[row * SX + col] += (acc[j][r] + bias) * gt[row * 8 + e];
          }
        }
      }
      __syncthreads();
    }
  }

  // output = x[:, -1, :]  (token 80)
  if (tid < DIM) out[b * DIM + tid] = xs[80 * SX + tid];
}

extern "C" void kernel_launch(void* const* d_in, const int* in_sizes, int n_in,
                              void* d_out, int out_size, void* d_ws, size_t ws_size,
                              hipStream_t stream) {
  (void)in_sizes; (void)n_in; (void)d_ws; (void)ws_size; (void)out_size;
  const int*   x_cat         = (const int*)  d_in[0];
  const float* x_num         = (const float*)d_in[1];
  const float* x_eng         = (const float*)d_in[2];
  const float* emb_table     = (const float*)d_in[3];
  const float* emb_eng_table = (const float*)d_in[4];
  const float* emb_bias      = (const float*)d_in[5];
  const float* eng_bias      = (const float*)d_in[6];
  const float* ln_eng_g      = (const float*)d_in[7];
  const float* ln_eng_b      = (const float*)d_in[8];
  const float* ln1_g         = (const float*)d_in[9];
  const float* ln1_b         = (const float*)d_in[10];
  const float* ln2_g         = (const float*)d_in[11];
  const float* ln2_b         = (const float*)d_in[12];
  const float* attn_in_w     = (const float*)d_in[13];
  const float* attn_in_b     = (const float*)d_in[14];
  const float* attn_out_w    = (const float*)d_in[15];
  const float* attn_out_b    = (const float*)d_in[16];
  const float* gate_w        = (const float*)d_in[17];
  const float* gate_b        = (const float*)d_in[18];
  const float* moe_w1        = (const float*)d_in[19];
  const float* moe_b1        = (const float*)d_in[20];
  const float* moe_w2        = (const float*)d_in[21];
  const float* moe_b2        = (const float*)d_in[22];
  float* out = (float*)d_out;

  size_t shmem = (size_t)(5 * S_PAD * SX + S_PAD * SS) * sizeof(float); // ~169 KB
  fam_forward<<<dim3(B_SZ), dim3(NTHREADS), shmem, stream>>>(
      x_cat, x_num, x_eng, emb_table, emb_eng_table, emb_bias, eng_bias,
      ln_eng_g, ln_eng_b, ln1_g, ln1_b, ln2_g, ln2_b,
      attn_in_w, attn_in_b, attn_out_w, attn_out_b,
      gate_w, gate_b, moe_w1, moe_b1, moe_w2, moe_b2, out);
}